// SMPL_19215683682772
// MI455X (gfx1250) — compile-verified
//
#include <hip/hip_runtime.h>

#define NJ   24
#define NPF  207
#define NXJ  9
#define NOUT 49

typedef __attribute__((ext_vector_type(2))) float v2f;
typedef __attribute__((ext_vector_type(8))) float v8f;

// ---------------------------------------------------------------------------
// K0: rotmat -> axis-angle -> rotmat (Rodrigues), pose_feature = R[1:]-I
// ---------------------------------------------------------------------------
__global__ void pose_kernel(const float* __restrict__ global_orient,
                            const float* __restrict__ body_pose,
                            float* __restrict__ R_out,     // [24*9]
                            float* __restrict__ pf_out) {  // [207]
  int j = threadIdx.x;
  if (j >= NJ) return;
  float Rm[9];
  const float* src = (j == 0) ? global_orient : (body_pose + (j - 1) * 9);
#pragma unroll
  for (int i = 0; i < 9; ++i) Rm[i] = src[i];
  // rotation_matrix_to_angle_axis
  float tr = Rm[0] + Rm[4] + Rm[8];
  float cosang = (tr - 1.0f) * 0.5f;
  cosang = fminf(fmaxf(cosang, -1.0f + 1e-6f), 1.0f - 1e-6f);
  float ang = acosf(cosang);
  float rx = Rm[7] - Rm[5];
  float ry = Rm[2] - Rm[6];
  float rz = Rm[3] - Rm[1];
  float inv = ang / (2.0f * sinf(ang) + 1e-8f);
  float ax = rx * inv, ay = ry * inv, az = rz * inv;
  // Rodrigues
  float theta = sqrtf(ax * ax + ay * ay + az * az);
  float Ro[9];
  if (theta < 1e-8f) {
    Ro[0] = 1.f; Ro[1] = 0.f; Ro[2] = 0.f;
    Ro[3] = 0.f; Ro[4] = 1.f; Ro[5] = 0.f;
    Ro[6] = 0.f; Ro[7] = 0.f; Ro[8] = 1.f;
  } else {
    float x = ax / theta, y = ay / theta, z = az / theta;
    float c = cosf(theta), s = sinf(theta), C = 1.0f - c;
    Ro[0] = c + x * x * C;   Ro[1] = x * y * C - z * s; Ro[2] = x * z * C + y * s;
    Ro[3] = y * x * C + z * s; Ro[4] = c + y * y * C;   Ro[5] = y * z * C - x * s;
    Ro[6] = z * x * C - y * s; Ro[7] = z * y * C + x * s; Ro[8] = c + z * z * C;
  }
#pragma unroll
  for (int i = 0; i < 9; ++i) R_out[j * 9 + i] = Ro[i];
  if (j >= 1) {
#pragma unroll
    for (int i = 0; i < 9; ++i) {
      float id = (i == 0 || i == 4 || i == 8) ? 1.0f : 0.0f;
      pf_out[(j - 1) * 9 + i] = Ro[i] - id;
    }
  }
}

// ---------------------------------------------------------------------------
// K1: v_shaped = v_template + shapedirs@betas ; v_posed = v_shaped + posedirs@pf
// Dominant-traffic kernel (posedirs = V*3*207 floats, contiguous per vertex).
// ---------------------------------------------------------------------------
__global__ void vertex_kernel(const float* __restrict__ betas,
                              const float* __restrict__ v_template,
                              const float* __restrict__ shapedirs,
                              const float* __restrict__ posedirs,
                              const float* __restrict__ pf_ws,
                              float* __restrict__ vshaped,
                              float* __restrict__ vposed, int V) {
  __shared__ float pf[NPF];
  __shared__ float bet[10];
  for (int i = threadIdx.x; i < NPF; i += blockDim.x) pf[i] = pf_ws[i];
  if (threadIdx.x < 10) bet[threadIdx.x] = betas[threadIdx.x];
  __syncthreads();
  int v = blockIdx.x * blockDim.x + threadIdx.x;
  if (v >= V) return;
  const float* sdv = shapedirs + (size_t)v * 30;
  const float* pdv = posedirs + (size_t)v * (3 * NPF);
  float vs[3];
#pragma unroll
  for (int k = 0; k < 3; ++k) {
    float a = v_template[3 * v + k];
#pragma unroll
    for (int b = 0; b < 10; ++b) a += sdv[k * 10 + b] * bet[b];
    vs[k] = a;
    vshaped[3 * v + k] = a;
  }
#pragma unroll
  for (int k = 0; k < 3; ++k) {
    const float* row = pdv + k * NPF;
    float acc = 0.f;
    for (int p = 0; p < NPF; ++p) acc += row[p] * pf[p];
    vposed[3 * v + k] = vs[k] + acc;
  }
}

// ---------------------------------------------------------------------------
// K2/K5: out[row,:] = sum_v reg[row*V+v] * vecs[v,:]   (one block per row)
// ---------------------------------------------------------------------------
__global__ void reduce_rows(const float* __restrict__ reg,
                            const float* __restrict__ vecs,
                            float* __restrict__ out, int V) {
  __shared__ float s[3 * 256];
  int row = blockIdx.x;
  const float* r = reg + (size_t)row * V;
  float ax = 0.f, ay = 0.f, az = 0.f;
  for (int v = threadIdx.x; v < V; v += blockDim.x) {
    float w = r[v];
    ax += w * vecs[3 * v + 0];
    ay += w * vecs[3 * v + 1];
    az += w * vecs[3 * v + 2];
  }
  s[threadIdx.x] = ax;
  s[256 + threadIdx.x] = ay;
  s[512 + threadIdx.x] = az;
  __syncthreads();
  for (int off = 128; off > 0; off >>= 1) {
    if ((int)threadIdx.x < off) {
      s[threadIdx.x] += s[threadIdx.x + off];
      s[256 + threadIdx.x] += s[256 + threadIdx.x + off];
      s[512 + threadIdx.x] += s[512 + threadIdx.x + off];
    }
    __syncthreads();
  }
  if (threadIdx.x == 0) {
    out[3 * row + 0] = s[0];
    out[3 * row + 1] = s[256];
    out[3 * row + 2] = s[512];
  }
}

// ---------------------------------------------------------------------------
// K3: kinematic chain (serial, tiny) -> G[24][16] row-major 4x4
// ---------------------------------------------------------------------------
__global__ void chain_kernel(const float* __restrict__ Rws,
                             const float* __restrict__ Jws,
                             const int* __restrict__ parents,
                             float* __restrict__ Gout) {
  if (threadIdx.x != 0) return;
  float A[NJ][16];
  float G[NJ][16];
  for (int j = 0; j < NJ; ++j) {
    float t0, t1, t2;
    if (j == 0) {
      t0 = Jws[0]; t1 = Jws[1]; t2 = Jws[2];
    } else {
      int p = parents[j];
      t0 = Jws[3 * j + 0] - Jws[3 * p + 0];
      t1 = Jws[3 * j + 1] - Jws[3 * p + 1];
      t2 = Jws[3 * j + 2] - Jws[3 * p + 2];
    }
    const float* R = Rws + j * 9;
    A[j][0] = R[0]; A[j][1] = R[1]; A[j][2]  = R[2]; A[j][3]  = t0;
    A[j][4] = R[3]; A[j][5] = R[4]; A[j][6]  = R[5]; A[j][7]  = t1;
    A[j][8] = R[6]; A[j][9] = R[7]; A[j][10] = R[8]; A[j][11] = t2;
    A[j][12] = 0.f; A[j][13] = 0.f; A[j][14] = 0.f;  A[j][15] = 1.f;
  }
  for (int e = 0; e < 16; ++e) G[0][e] = A[0][e];
  for (int j = 1; j < NJ; ++j) {
    int p = parents[j];
    for (int a = 0; a < 4; ++a)
      for (int b = 0; b < 4; ++b) {
        float acc = 0.f;
        for (int k = 0; k < 4; ++k) acc += G[p][a * 4 + k] * A[j][k * 4 + b];
        G[j][a * 4 + b] = acc;
      }
  }
  // subtract rest-pose joint contribution: G[:, :3, 3] -= G[:, :3, :3] @ J
  for (int j = 0; j < NJ; ++j) {
    for (int a = 0; a < 3; ++a) {
      float corr = G[j][a * 4 + 0] * Jws[3 * j + 0] +
                   G[j][a * 4 + 1] * Jws[3 * j + 1] +
                   G[j][a * 4 + 2] * Jws[3 * j + 2];
      G[j][a * 4 + 3] -= corr;
    }
    for (int e = 0; e < 16; ++e) Gout[j * 16 + e] = G[j][e];
  }
}

// ---------------------------------------------------------------------------
// K4: LBS with V_WMMA_F32_16X16X4_F32.
// T[v, e] = sum_j weights[v,j] * G[j,e]  -> 16 verts x 16 elems per wave,
// K = 24 joints in 6 chunks of 4. Then per-vertex affine transform.
// A layout (32-bit 16x4): lanes 0-15: VGPR0=K0,VGPR1=K1; lanes 16-31: K2,K3.
// B layout (4x16):        lanes 0-15: VGPR0=K0 row;      lanes 16-31: K2 row.
// C/D layout: VGPR r = row (8*laneHalf + r), col = lane&15.
// ---------------------------------------------------------------------------
__global__ void skin_kernel(const float* __restrict__ weights,
                            const float* __restrict__ Gws,
                            const float* __restrict__ vposed,
                            float* __restrict__ verts, int V) {
  __shared__ float ldsG[NJ * 16];   // 384
  __shared__ float ldsT[8 * 256];   // 8 waves * (16 verts x 16 elems)
  for (int i = threadIdx.x; i < NJ * 16; i += blockDim.x) ldsG[i] = Gws[i];
  __syncthreads();

  const int lane = threadIdx.x & 31;
  const int wave = threadIdx.x >> 5;
  const int laneHalf = lane >> 4;   // 0 or 1
  const int n = lane & 15;          // column (G element) / row M within tile
  const int vbase = blockIdx.x * 128 + wave * 16;

  int va = vbase + n;               // vertex providing this lane's A rows
  if (va > V - 1) va = V - 1;       // clamp (no divergence; EXEC stays full)
  const float* wrow = weights + (size_t)va * NJ + 2 * laneHalf;

  v8f acc;
#pragma unroll
  for (int i = 0; i < 8; ++i) acc[i] = 0.0f;

#pragma unroll
  for (int c = 0; c < 6; ++c) {
    v2f a, b;
    a.x = wrow[4 * c + 0];
    a.y = wrow[4 * c + 1];
    int kk = 4 * c + 2 * laneHalf;
    b.x = ldsG[(kk + 0) * 16 + n];
    b.y = ldsG[(kk + 1) * 16 + n];
    acc = __builtin_amdgcn_wmma_f32_16x16x4_f32(
        /*neg_a=*/false, a, /*neg_b=*/false, b,
        /*c_mod=*/(short)0, acc, /*reuse_a=*/false, /*reuse_b=*/false);
  }

  float* tb = ldsT + wave * 256;
#pragma unroll
  for (int r = 0; r < 8; ++r) tb[(laneHalf * 8 + r) * 16 + n] = acc[r];
  __syncthreads();

  if (threadIdx.x < 128) {
    int v = blockIdx.x * 128 + threadIdx.x;
    if (v < V) {
      const float* T = ldsT + threadIdx.x * 16;
      float px = vposed[3 * v + 0];
      float py = vposed[3 * v + 1];
      float pz = vposed[3 * v + 2];
      verts[3 * v + 0] = T[0] * px + T[1] * py + T[2]  * pz + T[3];
      verts[3 * v + 1] = T[4] * px + T[5] * py + T[6]  * pz + T[7];
      verts[3 * v + 2] = T[8] * px + T[9] * py + T[10] * pz + T[11];
    }
  }
}

// ---------------------------------------------------------------------------
// K6: joint_map gather
// ---------------------------------------------------------------------------
__global__ void map_kernel(const float* __restrict__ jointsAll,
                           const int* __restrict__ jmap,
                           float* __restrict__ out) {
  int i = threadIdx.x;
  if (i < NOUT) {
    int s = jmap[i];
    out[3 * i + 0] = jointsAll[3 * s + 0];
    out[3 * i + 1] = jointsAll[3 * s + 1];
    out[3 * i + 2] = jointsAll[3 * s + 2];
  }
}

extern "C" void kernel_launch(void* const* d_in, const int* in_sizes, int n_in,
                              void* d_out, int out_size, void* d_ws, size_t ws_size,
                              hipStream_t stream) {
  const float* betas         = (const float*)d_in[0];
  const float* body_pose     = (const float*)d_in[1];
  const float* global_orient = (const float*)d_in[2];
  const float* v_template    = (const float*)d_in[3];
  const float* shapedirs     = (const float*)d_in[4];
  const float* posedirs      = (const float*)d_in[5];
  const float* Jreg          = (const float*)d_in[6];
  const float* JregX         = (const float*)d_in[7];
  const float* weights       = (const float*)d_in[8];
  const int*   parents       = (const int*)d_in[9];
  const int*   joint_map     = (const int*)d_in[10];
  const int V = in_sizes[3] / 3;

  float* ws        = (float*)d_ws;
  float* R_ws      = ws;          // 216
  float* pf_ws     = ws + 216;    // 207
  float* J_ws      = ws + 448;    // 72
  float* G_ws      = ws + 520;    // 384
  float* joints_ws = ws + 904;    // 99 (24 regular + 9 extra)
  float* vshaped   = ws + 1024;   // 3V
  float* vposed    = vshaped + (size_t)3 * V;

  float* verts      = (float*)d_out;          // [V,3]
  float* out_joints = verts + (size_t)3 * V;  // [49,3]

  pose_kernel<<<1, 32, 0, stream>>>(global_orient, body_pose, R_ws, pf_ws);

  int nb1 = (V + 255) / 256;
  vertex_kernel<<<nb1, 256, 0, stream>>>(betas, v_template, shapedirs, posedirs,
                                         pf_ws, vshaped, vposed, V);

  reduce_rows<<<NJ, 256, 0, stream>>>(Jreg, vshaped, J_ws, V);

  chain_kernel<<<1, 32, 0, stream>>>(R_ws, J_ws, parents, G_ws);

  int nb4 = (V + 127) / 128;
  skin_kernel<<<nb4, 256, 0, stream>>>(weights, G_ws, vposed, verts, V);

  reduce_rows<<<NJ, 256, 0, stream>>>(Jreg, verts, joints_ws, V);
  reduce_rows<<<NXJ, 256, 0, stream>>>(JregX, verts, joints_ws + 3 * NJ, V);

  map_kernel<<<1, 64, 0, stream>>>(joints_ws, joint_map, out_joints);
}